// CoPE_37830071943369
// MI455X (gfx1250) — compile-verified
//
#include <hip/hip_runtime.h>
#include <hip/hip_bf16.h>

typedef __attribute__((ext_vector_type(2))) float v2f;
typedef __attribute__((ext_vector_type(8))) float v8f;

#define CB 2
#define CH 16
#define CS 2048
#define CD 64
#define CNPOS 64
#define NROWS (CB * CH * CS)   // 65536 query rows

// ---------------------------------------------------------------------------
// Kernel 1: logits_int[r, p] = sum_d query[r, d] * pos_emb[d, p]
// fp32 WMMA 16x16x4. Each wave32 computes a 16-row x 64-col tile:
//   4 accumulators (N tiles of 16) x 16 K-steps (K=64 in chunks of 4).
//
// Fragment layouts (CDNA5 ISA 7.12.2, 32-bit data):
//   A 16x4 : lanes 0-15 -> M=lane, VGPR{0,1}=K{0,1}; lanes 16-31 -> K{2,3}
//   B 4x16 : lanes 0-15 -> N=lane, VGPR{0,1}=K{0,1}; lanes 16-31 -> K{2,3}
//   C 16x16: VGPR i -> (M=i, N=lane) for lanes 0-15, (M=8+i, N=lane-16) else
// ---------------------------------------------------------------------------
__global__ __launch_bounds__(128) void cope_gemm_wmma(
    const float* __restrict__ Q,   // [NROWS, 64]
    const float* __restrict__ P,   // [64, 64]  (d-major, pos minor)
    float* __restrict__ L)         // [NROWS, 64]
{
  const int wave = threadIdx.x >> 5;
  const int lane = threadIdx.x & 31;
  const int tile = blockIdx.x * 4 + wave;          // 16-row tile index
  const long row0 = (long)tile * 16;
  if (row0 >= NROWS) return;

  const int half = lane >> 4;                      // 0: K pair {0,1}, 1: {2,3}
  const int l16  = lane & 15;

  v8f acc0 = {}, acc1 = {}, acc2 = {}, acc3 = {};

  const float* qrow = Q + (row0 + l16) * CD;

#pragma unroll
  for (int k = 0; k < 16; ++k) {
    const int kk = k * 4 + half * 2;               // base K of my pair
    v2f a;
    a.x = qrow[kk];
    a.y = qrow[kk + 1];

    const float* p0 = P + (size_t)kk * CNPOS;      // row K=kk of pos_emb
    const float* p1 = p0 + CNPOS;                  // row K=kk+1
    v2f b0, b1, b2, b3;
    b0.x = p0[l16];       b0.y = p1[l16];
    b1.x = p0[16 + l16];  b1.y = p1[16 + l16];
    b2.x = p0[32 + l16];  b2.y = p1[32 + l16];
    b3.x = p0[48 + l16];  b3.y = p1[48 + l16];

    acc0 = __builtin_amdgcn_wmma_f32_16x16x4_f32(false, a, false, b0, (short)0, acc0, false, false);
    acc1 = __builtin_amdgcn_wmma_f32_16x16x4_f32(false, a, false, b1, (short)0, acc1, false, false);
    acc2 = __builtin_amdgcn_wmma_f32_16x16x4_f32(false, a, false, b2, (short)0, acc2, false, false);
    acc3 = __builtin_amdgcn_wmma_f32_16x16x4_f32(false, a, false, b3, (short)0, acc3, false, false);
  }

  float* obase = L + row0 * CNPOS;
  const int col = l16;
#pragma unroll
  for (int i = 0; i < 8; ++i) {
    const int r = i + half * 8;
    obase[(size_t)r * CNPOS + col]      = acc0[i];
    obase[(size_t)r * CNPOS + 16 + col] = acc1[i];
    obase[(size_t)r * CNPOS + 32 + col] = acc2[i];
    obase[(size_t)r * CNPOS + 48 + col] = acc3[i];
  }
}

// ---------------------------------------------------------------------------
// Kernel 2: per output row (b,h,s):
//   gates = sigmoid(attn_logits[row, :])            (2048 elems)
//   pos   = reverse cumsum(gates), clipped at 63
//   out   = lerp(logits_int[row, floor], logits_int[row, ceil], frac)
// One 256-thread block per row; 8 contiguous elements per thread (2x float4).
// Block-wide SUFFIX scan of per-thread chunk sums via Hillis-Steele in LDS.
// ---------------------------------------------------------------------------
__global__ __launch_bounds__(256) void cope_main(
    const float* __restrict__ A,   // attn_logits [NROWS, 2048]
    const float* __restrict__ L,   // logits_int  [NROWS, 64]
    float* __restrict__ O)         // out         [NROWS, 2048]
{
  const int EPT = 8;
  const int row = blockIdx.x;
  const int t   = threadIdx.x;

  const float* __restrict__ arow = A + (size_t)row * CS;
  float* __restrict__ orow       = O + (size_t)row * CS;

  __shared__ float sLog[CNPOS];
  __shared__ float sSum[256];

  if (t < CNPOS) sLog[t] = L[(size_t)row * CNPOS + t];

  // -- load 8 gates (two b128 loads), apply sigmoid --------------------------
  const int base = t * EPT;
  float g[EPT];
  {
    const float4* a4 = (const float4*)(arow + base);
    float4 v0 = a4[0];
    float4 v1 = a4[1];
    g[0] = v0.x; g[1] = v0.y; g[2] = v0.z; g[3] = v0.w;
    g[4] = v1.x; g[5] = v1.y; g[6] = v1.z; g[7] = v1.w;
  }
#pragma unroll
  for (int i = 0; i < EPT; ++i)
    g[i] = 1.0f / (1.0f + __expf(-g[i]));

  float csum = 0.0f;
#pragma unroll
  for (int i = 0; i < EPT; ++i) csum += g[i];

  sSum[t] = csum;
  __syncthreads();

  // -- inclusive SUFFIX scan over 256 chunk sums -----------------------------
#pragma unroll
  for (int off = 1; off < 256; off <<= 1) {
    float mine = sSum[t];
    float add  = (t + off < 256) ? sSum[t + off] : 0.0f;
    __syncthreads();
    sSum[t] = mine + add;
    __syncthreads();
  }
  const float suffixExcl = sSum[t] - csum;   // sum of all chunks strictly after mine

  // -- per-element reverse cumsum + gather/lerp ------------------------------
  float res[EPT];
  float run = suffixExcl;
#pragma unroll
  for (int i = EPT - 1; i >= 0; --i) {
    run += g[i];
    float pos = fminf(run, (float)(CNPOS - 1));
    float flf = floorf(pos);
    int   fl  = (int)flf;
    int   ce  = (int)ceilf(pos);
    float w   = pos - flf;
    float lf  = sLog[fl];
    float lc  = sLog[ce];
    res[i] = lc * w + lf * (1.0f - w);
  }

  {
    float4* o4 = (float4*)(orow + base);
    o4[0] = make_float4(res[0], res[1], res[2], res[3]);
    o4[1] = make_float4(res[4], res[5], res[6], res[7]);
  }
}

// ---------------------------------------------------------------------------
extern "C" void kernel_launch(void* const* d_in, const int* in_sizes, int n_in,
                              void* d_out, int out_size, void* d_ws, size_t ws_size,
                              hipStream_t stream) {
  const float* query   = (const float*)d_in[0];   // [B,H,S,D]
  const float* attn    = (const float*)d_in[1];   // [B,H,S,S]
  const float* pos_emb = (const float*)d_in[2];   // [1,D,NPOS]
  float* out           = (float*)d_out;           // [B,H,S,S]
  float* logits_int    = (float*)d_ws;            // [NROWS, NPOS] = 16 MB scratch

  // Kernel 1: 65536 rows / 16 rows-per-wave / 4 waves-per-block = 1024 blocks
  dim3 g1(NROWS / 16 / 4), b1(128);
  cope_gemm_wmma<<<g1, b1, 0, stream>>>(query, pos_emb, logits_int);

  // Kernel 2: one block per output row
  dim3 g2(NROWS), b2(256);
  cope_main<<<g2, b2, 0, stream>>>(attn, logits_int, out);
}